// pairwise_ranking_objective_59691455480291
// MI455X (gfx1250) — compile-verified
//
#include <hip/hip_runtime.h>
#include <math.h>
#include <stdint.h>

typedef __attribute__((ext_vector_type(2))) float v2f;
typedef __attribute__((ext_vector_type(8))) float v8f;

#define MARGIN      0.7f
#define EPSN        1e-12f
#define DIMS        256
#define ROWS_PG     16            // rows per wave-group (WMMA M)
#define PHASE_COLS  32            // K columns staged per phase
#define NPHASES     (DIMS / PHASE_COLS)          // 8
#define ROW_STRIDE  36            // dwords; 36L mod 64 distinct, halves disjoint mod 4
#define BUF_DW      (ROWS_PG * ROW_STRIDE)       // 576 dwords per buffer
#define WAVES_PB    8
#define MOFF        (WAVES_PB * 2 * BUF_DW)      // 9216: m_hat region
#define WSOFF       (MOFF + DIMS)                // 9472: per-wave sums
#define SMEM_DW     (WSOFF + WAVES_PB)           // 9480 dwords = 37920 B

// ---------------------------------------------------------------------------
// Kernel 1: one wave. Normalize midi, compute s_match.
// ws[0..255] = m_hat, ws[256] = s_match
// ---------------------------------------------------------------------------
__global__ void prb_setup_kernel(const float* __restrict__ midi,
                                 const float* __restrict__ audio,
                                 float* __restrict__ ws) {
    const int lane = threadIdx.x;            // 0..31, wave32
    float mm = 0.f, aa = 0.f, ma = 0.f;
    float mv[8];
#pragma unroll
    for (int i = 0; i < 8; ++i) {
        float m = midi[lane + 32 * i];
        float a = audio[lane + 32 * i];
        mv[i] = m;
        mm = fmaf(m, m, mm);
        aa = fmaf(a, a, aa);
        ma = fmaf(m, a, ma);
    }
#pragma unroll
    for (int off = 16; off >= 1; off >>= 1) {
        mm += __shfl_xor(mm, off);
        aa += __shfl_xor(aa, off);
        ma += __shfl_xor(ma, off);
    }
    const float nm = fmaxf(sqrtf(mm), EPSN);
    const float na = fmaxf(sqrtf(aa), EPSN);
    const float inv_m = 1.0f / nm;
    const float s_match = ma / (nm * na);
#pragma unroll
    for (int i = 0; i < 8; ++i) ws[lane + 32 * i] = mv[i] * inv_m;
    if (lane == 0) ws[DIMS] = s_match;
}

// Issue one phase's async global->LDS copies: 4 x b128 per lane
// (16 rows x 32 cols, 128B-burst coalesced; rows 4j..4j+3 per instruction).
__device__ __forceinline__ void issue_phase_async(const float* gptr, uint32_t lbyte) {
#pragma unroll
    for (int j = 0; j < 4; ++j) {
        asm volatile("global_load_async_to_lds_b128 %0, %1, off"
                     :: "v"(lbyte + (uint32_t)(j * 4 * ROW_STRIDE * 4)),
                        "v"((uint64_t)(uintptr_t)(gptr + j * 4 * DIMS))
                     : "memory");
    }
}

// ---------------------------------------------------------------------------
// Kernel 2: WMMA hinge reduction with double-buffered async LDS staging.
// ---------------------------------------------------------------------------
__global__ void prb_hinge_wmma_kernel(const float* __restrict__ C,
                                      const float* __restrict__ ws,
                                      float* __restrict__ partial) {
    __shared__ float smem[SMEM_DW];

    const int tid  = threadIdx.x;
    const int lane = tid & 31;
    const int wave = tid >> 5;

    // stage normalized midi into LDS (blockDim == 256 == DIMS)
    smem[MOFF + tid] = ws[tid];
    const float s_match = ws[DIMS];

    const int group = blockIdx.x * WAVES_PB + wave;
    const float* src = C + (size_t)group * (ROWS_PG * DIMS);

    // async staging addressing: lane -> (row r0 = lane>>3 within 4-row step, f4 col)
    const int r0 = lane >> 3;
    const int c4 = lane & 7;
    const float* gbase = src + r0 * DIMS + c4 * 4;
    float* buf0 = smem + wave * (2 * BUF_DW);
    // LDS byte address: low 32 bits of flat shared address == LDS offset
    const uint32_t lbase =
        (uint32_t)(uintptr_t)buf0 + (uint32_t)((r0 * ROW_STRIDE + c4 * 4) * 4);

    // prefetch phase 0 into buffer 0 while block finishes m_hat staging
    issue_phase_async(gbase, lbase);
    __syncthreads();

    const int sub  = (lane >> 4) << 1;   // 0 (lanes 0-15) or 2 (lanes 16-31)
    const int rowA = lane & 15;          // A-frag row this lane reads

    v8f acc = {0.f, 0.f, 0.f, 0.f, 0.f, 0.f, 0.f, 0.f};
    float nrm = 0.f;

#pragma unroll 1
    for (int p = 0; p < NPHASES; ++p) {
        if (p < NPHASES - 1) {
            // double-buffer: start next phase's copies, then wait for current
            issue_phase_async(gbase + (p + 1) * PHASE_COLS,
                              lbase + (uint32_t)(((p + 1) & 1) * BUF_DW * 4));
            asm volatile("s_wait_asynccnt 0x4" ::: "memory");
        } else {
            asm volatile("s_wait_asynccnt 0x0" ::: "memory");
        }

        const float* aRow = smem + wave * (2 * BUF_DW) + (p & 1) * BUF_DW
                          + rowA * ROW_STRIDE;
        const float* mPh  = smem + MOFF + p * PHASE_COLS;
#pragma unroll
        for (int k0 = 0; k0 < PHASE_COLS; k0 += 4) {
            v2f af = *(const v2f*)(aRow + k0 + sub);   // A: c[rowA, k0+sub..+1]
            v2f bm = *(const v2f*)(mPh + k0 + sub);    // B: m_hat broadcast over N
            nrm = fmaf(af.x, af.x, nrm);               // row-norm partial on VALU
            nrm = fmaf(af.y, af.y, nrm);
            acc = __builtin_amdgcn_wmma_f32_16x16x4_f32(
                false, af, false, bm, (short)0, acc, false, false);
        }
    }

    // ---- epilogue ----
    // nrm(lane) covers row (lane&15), k = sub,sub+1 (mod 4); pair with lane^16
    const float nrm_full = nrm + __shfl_xor(nrm, 16);     // ||c_{lane&15}||^2

    // acc[v] at lane l holds D[M][N], M = v + 8*(l>=16) (same value for all N)
    float t = 0.f;
#pragma unroll
    for (int v = 0; v < 8; ++v) t = ((lane & 7) == v) ? acc[v] : t;
    // t(lane) = dot(row (lane&7) + 8*(lane>=16)); gather row (lane&15)
    const int lr = lane & 15;
    const float dotv = __shfl(t, (lr < 8) ? lr : (lr + 8));

    float contrib = 0.f;
    if (lane < 16) {
        const float cosv = dotv / fmaxf(sqrtf(nrm_full), EPSN);
        contrib = fmaxf(0.f, MARGIN - s_match + cosv);
    }
#pragma unroll
    for (int off = 16; off >= 1; off >>= 1) contrib += __shfl_xor(contrib, off);
    if (lane == 0) smem[WSOFF + wave] = contrib;
    __syncthreads();
    if (tid == 0) {
        float s = 0.f;
#pragma unroll
        for (int w = 0; w < WAVES_PB; ++w) s += smem[WSOFF + w];
        partial[blockIdx.x] = s;
    }
}

// ---------------------------------------------------------------------------
// Kernel 3: one wave reduces the per-block partials into d_out[0].
// ---------------------------------------------------------------------------
__global__ void prb_final_kernel(const float* __restrict__ partial,
                                 float* __restrict__ out, int nblocks) {
    const int lane = threadIdx.x;
    float s = 0.f;
    for (int i = lane; i < nblocks; i += 32) s += partial[i];
#pragma unroll
    for (int off = 16; off >= 1; off >>= 1) s += __shfl_xor(s, off);
    if (lane == 0) out[0] = s;
}

extern "C" void kernel_launch(void* const* d_in, const int* in_sizes, int n_in,
                              void* d_out, int out_size, void* d_ws, size_t ws_size,
                              hipStream_t stream) {
    const float* midi  = (const float*)d_in[0];   // (1,256) f32
    const float* audio = (const float*)d_in[1];   // (1,256) f32
    const float* C     = (const float*)d_in[2];   // (N,256) f32
    float* ws  = (float*)d_ws;
    float* out = (float*)d_out;

    const int N = in_sizes[2] / DIMS;                 // 131072
    const int nblocks = N / (WAVES_PB * ROWS_PG);     // 1024 (exact)

    prb_setup_kernel<<<1, 32, 0, stream>>>(midi, audio, ws);
    prb_hinge_wmma_kernel<<<nblocks, 256, 0, stream>>>(C, ws, ws + 512);
    prb_final_kernel<<<1, 32, 0, stream>>>(ws + 512, out, nblocks);
}